// QuantIntSoft_75462575391085
// MI455X (gfx1250) — compile-verified
//
#include <hip/hip_runtime.h>

#define N_SHIFT 30
#define X0f     (-0.6931f)
#define COEF0f  0.35815147f
#define COEF1f  0.96963238f
#define ROW_LEN 2048
#define BLOCK   256
#define WAVES   (BLOCK / 32)

typedef __attribute__((address_space(3))) float lds_float;

__device__ __forceinline__ float wave_max(float v) {
#pragma unroll
  for (int m = 16; m >= 1; m >>= 1) v = fmaxf(v, __shfl_xor(v, m, 32));
  return v;
}
__device__ __forceinline__ float wave_sum(float v) {
#pragma unroll
  for (int m = 16; m >= 1; m >>= 1) v += __shfl_xor(v, m, 32);
  return v;
}

__global__ __launch_bounds__(BLOCK) void QuantIntSoft_75462575391085_kernel(
    const float* __restrict__ x, const float* __restrict__ scale,
    float* __restrict__ out) {
  __shared__ float buf[ROW_LEN];   // 8 KB row staging
  __shared__ float red[WAVES];

  const int tid = threadIdx.x;
  const long long row = blockIdx.x;
  const float* grow = x + row * (long long)ROW_LEN;
  float* orow = out + row * (long long)ROW_LEN;

  // ---- gfx1250 async DMA: issue the 8KB row copy into LDS ----
  // Each lane copies 16B at byte offset tid*16, twice (offset 0 and 4096).
  // INST_OFFSET is added to BOTH the LDS and global addresses (ISA 08 §4.4).
  unsigned lds_addr = (unsigned)(unsigned long long)(lds_float*)buf + tid * 16u;
  const float* g = grow + tid * 4;
  asm volatile(
      "global_load_async_to_lds_b128 %0, %1, off\n\t"
      "global_load_async_to_lds_b128 %0, %1, off offset:4096"
      :
      : "v"(lds_addr), "v"(g)
      : "memory");

  // ---- uniform scalar setup overlaps with the in-flight async DMA ----
  const float s      = scale[0];
  const float inv_s  = 1.0f / s;
  const float x0_int = floorf(X0f / s);                       // negative
  const float b_int  = floorf((COEF1f / COEF0f) / s);
  const float c_int  = floorf((1.0f / COEF0f) / (s * s));
  const float lo     = (float)N_SHIFT * x0_int;
  const float inv_x0 = 1.0f / x0_int;

  // drain the DMA, then make LDS visible
  asm volatile("s_wait_asynccnt 0" ::: "memory");
  __syncthreads();

  // ---- pass 1: load 8 elems/thread from LDS, scale, row max ----
  const float4* b4 = (const float4*)buf;
  float4 va = b4[tid];
  float4 vb = b4[BLOCK + tid];
  float t[8] = {va.x, va.y, va.z, va.w, vb.x, vb.y, vb.z, vb.w};

  float m = -__builtin_inff();
#pragma unroll
  for (int i = 0; i < 8; ++i) {
    t[i] *= inv_s;
    m = fmaxf(m, t[i]);
  }
  m = wave_max(m);
  if ((tid & 31) == 0) red[tid >> 5] = m;
  __syncthreads();
  float tmax = red[0];
#pragma unroll
  for (int i = 1; i < WAVES; ++i) tmax = fmaxf(tmax, red[i]);
  __syncthreads();

  // ---- pass 2: int_exp + row sum ----
  float e[8];
  float esum = 0.0f;
#pragma unroll
  for (int i = 0; i < 8; ++i) {
    float xi = fmaxf(t[i] - tmax, lo);          // clamp at 30*x0_int
    float q  = floorf(xi * inv_x0);             // 0..30 integer
    float r  = xi - x0_int * q;
    float z  = r * (r + b_int) + c_int;
    float p  = exp2f((float)N_SHIFT - q);       // exact power of two
    float ev = fmaxf(floorf(z * p), 0.0f);
    e[i] = ev;
    esum += ev;
  }
  esum = wave_sum(esum);
  if ((tid & 31) == 0) red[tid >> 5] = esum;
  __syncthreads();
  float exp_sum = 0.0f;
#pragma unroll
  for (int i = 0; i < WAVES; ++i) exp_sum += red[i];
  __syncthreads();

  // ---- pass 3: inverse-softmax int, log2-round, dequant + row sum ----
  float d[8];
  float dsum = 0.0f;
#pragma unroll
  for (int i = 0; i < 8; ++i) {
    float v    = rintf(exp_sum / e[i]);         // e==0 -> inf (matches ref)
    float big  = floorf(log2f(v));
    float bump = ((v - exp2f(big)) >= exp2f(big - 1.0f)) ? 1.0f : 0.0f;
    float qlog = big + bump;                    // inf path: NaN cmp -> false -> inf
    qlog = fminf(fmaxf(qlog, 0.0f), 15.0f);     // clip to [0, 2^BITS-1]
    float dv = exp2f(-qlog);                    // exact
    d[i] = dv;
    dsum += dv;
  }
  dsum = wave_sum(dsum);
  if ((tid & 31) == 0) red[tid >> 5] = dsum;
  __syncthreads();
  float deq_sum = 0.0f;
#pragma unroll
  for (int i = 0; i < WAVES; ++i) deq_sum += red[i];

  // ---- pass 4: normalize + store ----
  const float inv_d = 1.0f / deq_sum;
  float4 r0 = make_float4(d[0] * inv_d, d[1] * inv_d, d[2] * inv_d, d[3] * inv_d);
  float4 r1 = make_float4(d[4] * inv_d, d[5] * inv_d, d[6] * inv_d, d[7] * inv_d);
  float4* o4 = (float4*)orow;
  o4[tid] = r0;
  o4[BLOCK + tid] = r1;
}

extern "C" void kernel_launch(void* const* d_in, const int* in_sizes, int n_in,
                              void* d_out, int out_size, void* d_ws, size_t ws_size,
                              hipStream_t stream) {
  const float* x     = (const float*)d_in[0];
  const float* scale = (const float*)d_in[1];
  float* out         = (float*)d_out;
  const int rows = in_sizes[0] / ROW_LEN;  // 16*2048 = 32768
  QuantIntSoft_75462575391085_kernel<<<rows, BLOCK, 0, stream>>>(x, scale, out);
}